// MSTE_62483184222722
// MI455X (gfx1250) — compile-verified
//
#include <hip/hip_runtime.h>
#include <cstddef>

// ---------------------------------------------------------------------------
// MI455X (gfx1250) implementation.
// Dominant cost: graph propagation GEMMs out = cur + 0.5*An@cur,
// An: 512x512 f32, cur: 512 x (256*L) f32, L in {20,14,8,2}.
// Done with V_WMMA_F32_16X16X4_F32 (full f32 precision, matrix pipe),
// 32x32 wave tiles (4 independent accumulators), L templated so all GEMM
// addressing is immediate-offset.  Everything is L2-resident (<70MB).
// ---------------------------------------------------------------------------

typedef __attribute__((ext_vector_type(2))) float v2f;
typedef __attribute__((ext_vector_type(8))) float v8f;

#define E_ 2
#define NB_ 8
#define B_ 8
#define N_ 512
#define SEQ_ 12
#define CIN_ 2
#define CONV_ 32
#define END_ 64
#define RF_ 26

// ----------------------------- utility kernels -----------------------------

__global__ void k_zero(float* __restrict__ p, int n) {
  int i = blockIdx.x * blockDim.x + threadIdx.x;
  if (i < n) p[i] = 0.0f;
}

__global__ void k_copy(const float* __restrict__ s, float* __restrict__ d, int n) {
  int i = blockIdx.x * blockDim.x + threadIdx.x;
  if (i < n) d[i] = s[i];
}

// ------------------------- adjacency normalization -------------------------

__global__ void k_deg(const float* __restrict__ adj, float* __restrict__ di) {
  __shared__ float red[256];
  int v = blockIdx.x;
  int tid = threadIdx.x;
  float s = 0.f;
  for (int w = tid; w < N_; w += 256) {
    float a = adj[v * N_ + w] + (w == v ? 1.0f : 0.0f);
    s += a;
  }
  red[tid] = s;
  __syncthreads();
  for (int off = 128; off > 0; off >>= 1) {
    if (tid < off) red[tid] += red[tid + off];
    __syncthreads();
  }
  if (tid == 0) di[v] = rsqrtf(red[0]);
}

__global__ void k_an(const float* __restrict__ adj, const float* __restrict__ di,
                     float* __restrict__ An, float* __restrict__ AnT) {
  int i = blockIdx.x * blockDim.x + threadIdx.x;
  if (i >= N_ * N_) return;
  int v = i >> 9, w = i & (N_ - 1);
  float a = adj[i] + (v == w ? 1.0f : 0.0f);
  float val = di[v] * a * di[w];
  An[v * N_ + w]  = val;
  AnT[w * N_ + v] = val;
}

// ------------------------------- start conv --------------------------------
__global__ void k_start(const float* __restrict__ bkc, float* __restrict__ x,
                        const float* __restrict__ Ws, const float* __restrict__ bs) {
  int idx = blockIdx.x * blockDim.x + threadIdx.x;
  const int TOT = B_ * CONV_ * N_ * RF_;
  if (idx >= TOT) return;
  int t = idx % RF_;
  int v = (idx / RF_) % N_;
  int o = (idx / (RF_ * N_)) % CONV_;
  int b = idx / (RF_ * N_ * CONV_);
  float val = bs[o];
  if (t >= RF_ - SEQ_) {
    int s = t - (RF_ - SEQ_);
    const float* p = bkc + (((size_t)b * SEQ_ + s) * N_ + v) * CIN_;
    val += Ws[o * 2] * p[0] + Ws[o * 2 + 1] * p[1];
  }
  x[idx] = val;
}

// --------------------------- gated inception -------------------------------
// h_nm[v][ b*32L + c*L + l ] = tanh(F)*sigmoid(G), node-major for the GEMM.
template <int L>
__global__ void k_inception(const float* __restrict__ x, float* __restrict__ h_nm,
                            const float* __restrict__ Wf2, const float* __restrict__ Wf3,
                            const float* __restrict__ Wf5, const float* __restrict__ Wf7,
                            const float* __restrict__ Wg2, const float* __restrict__ Wg3,
                            const float* __restrict__ Wg5, const float* __restrict__ Wg7,
                            const float* __restrict__ bF, const float* __restrict__ bG) {
  constexpr int CURL = L + 6;
  constexpr int NC = 256 * L;
  __shared__ float xs[CONV_ * RF_];
  int bv = blockIdx.x;
  int v = bv & (N_ - 1);
  int b = bv >> 9;
  int tid = threadIdx.x;
  constexpr int t0 = RF_ - CURL;
  for (int i = tid; i < CONV_ * CURL; i += 256) {
    int ic = i / CURL, t = i % CURL;
    xs[ic * RF_ + t] = x[(((size_t)b * CONV_ + ic) * N_ + v) * RF_ + t0 + t];
  }
  __syncthreads();
  size_t obase = (size_t)v * NC + (size_t)b * CONV_ * L;
  for (int idx = tid; idx < CONV_ * L; idx += 256) {
    int c = idx / L, l = idx % L;
    int g = c >> 3, oc = c & 7;
    const float* Wf; const float* Wg; int ksz;
    switch (g) {
      case 0:  Wf = Wf2; Wg = Wg2; ksz = 2; break;
      case 1:  Wf = Wf3; Wg = Wg3; ksz = 3; break;
      case 2:  Wf = Wf5; Wg = Wg5; ksz = 5; break;
      default: Wf = Wf7; Wg = Wg7; ksz = 7; break;
    }
    int off = 7 - ksz;                         // all windows end at l+6
    float F = bF[c], G = bG[c];
    for (int ic = 0; ic < CONV_; ++ic) {
      const float* xr = xs + ic * RF_ + l + off;
      const float* wf = Wf + (oc * CONV_ + ic) * ksz;
      const float* wg = Wg + (oc * CONV_ + ic) * ksz;
      for (int j = 0; j < ksz; ++j) {
        F += wf[j] * xr[j];
        G += wg[j] * xr[j];
      }
    }
    h_nm[obase + idx] = tanhf(F) * (1.0f / (1.0f + __expf(-G)));
  }
}

// ------------------- graph propagation GEMM (WMMA f32) ---------------------
// out(512 x NC) = src + 0.5 * M(512x512) @ src, all row-major f32.
// Block = 8 waves sharing a 32-row A panel in LDS (64KB); each wave owns a
// 32Mx32N tile = 4 independent f32 WMMA accumulators.  NC is compile-time,
// so B loads are immediate-offset off one incremented pointer.
template <int L>
__global__ void k_gprop(const float* __restrict__ Mm, const float* __restrict__ src,
                        float* __restrict__ out) {
  constexpr int NC = 256 * L;
  __shared__ __align__(16) float Apan[32 * N_];   // 64 KB
  const int mbase = blockIdx.y * 32;
  const int tid = threadIdx.x;
  for (int i = tid; i < (32 * N_) / 4; i += 256) {
    int row = i >> 7;              // 128 float4 per row
    int c4  = (i & 127) << 2;
    ((float4*)Apan)[i] = *(const float4*)(Mm + (size_t)(mbase + row) * N_ + c4);
  }
  __syncthreads();
  const int wave = tid >> 5;
  const int lane = tid & 31;
  const int nbase = blockIdx.x * 256 + wave * 32;
  const int mrow  = lane & 15;                 // A-frag row within panel half
  const int khalf = (lane >> 4) << 1;          // lanes 16-31 hold K+2,K+3
  const int ncol  = nbase + (lane & 15);       // B/C-frag column

  const float* pa = Apan + mrow * N_ + khalf;
  const float* pb = src + (size_t)khalf * NC + ncol;

  v8f acc00 = {}, acc01 = {}, acc10 = {}, acc11 = {};
#pragma unroll 2
  for (int k = 0; k < N_; k += 4) {
    v2f a0 = *(const v2f*)pa;                  // rows mbase..+15
    v2f a1 = *(const v2f*)(pa + 16 * N_);      // rows mbase+16..+31
    v2f b0, b1;
    b0.x = pb[0];
    b0.y = pb[NC];
    b1.x = pb[16];
    b1.y = pb[NC + 16];
    // WGP-scope prefetch of the B stream a few K-steps ahead (data is
    // consumed by this same wave -> high locality, near-cache scope).
    __builtin_prefetch(pb + 8 * NC, 0, 3);
    acc00 = __builtin_amdgcn_wmma_f32_16x16x4_f32(false, a0, false, b0,
                                                  (short)0, acc00, false, false);
    acc01 = __builtin_amdgcn_wmma_f32_16x16x4_f32(false, a0, false, b1,
                                                  (short)0, acc01, false, false);
    acc10 = __builtin_amdgcn_wmma_f32_16x16x4_f32(false, a1, false, b0,
                                                  (short)0, acc10, false, false);
    acc11 = __builtin_amdgcn_wmma_f32_16x16x4_f32(false, a1, false, b1,
                                                  (short)0, acc11, false, false);
    pa += 4;
    pb += 4 * NC;
  }
  const int rsel = (lane >> 4) << 3;           // lanes 16-31 hold rows M=8..15
#pragma unroll
  for (int r = 0; r < 8; ++r) {
    const int m0 = mbase + rsel + r;
    size_t i0 = (size_t)m0 * NC + ncol;
    size_t i1 = i0 + (size_t)16 * NC;
    out[i0]      = src[i0]      + 0.5f * acc00[r];
    out[i0 + 16] = src[i0 + 16] + 0.5f * acc01[r];
    out[i1]      = src[i1]      + 0.5f * acc10[r];
    out[i1 + 16] = src[i1 + 16] + 0.5f * acc11[r];
  }
}

// -------------- fused 96->32 channel mix (both dirs) + x update ------------
template <int L>
__global__ void k_mix(float* __restrict__ x,
                      const float* __restrict__ Hs, const float* __restrict__ S1,
                      const float* __restrict__ S2, const float* __restrict__ T1,
                      const float* __restrict__ T2,
                      const float* __restrict__ Wg1, const float* __restrict__ Wg2,
                      const float* __restrict__ bg1, const float* __restrict__ bg2) {
  constexpr int NC = 256 * L;
  constexpr int n = CONV_ * L;
  __shared__ float sl[5 * n];
  int bv = blockIdx.x;
  int v = bv & (N_ - 1);
  int b = bv >> 9;
  int tid = threadIdx.x;
  size_t base = (size_t)v * NC + (size_t)b * CONV_ * L;   // contiguous 32*L run
  for (int i = tid; i < n; i += 256) {
    sl[0 * n + i] = Hs[base + i];
    sl[1 * n + i] = S1[base + i];
    sl[2 * n + i] = S2[base + i];
    sl[3 * n + i] = T1[base + i];
    sl[4 * n + i] = T2[base + i];
  }
  __syncthreads();
  for (int idx = tid; idx < n; idx += 256) {
    int o = idx / L, l = idx % L;
    const float* w1 = Wg1 + o * 96;
    const float* w2 = Wg2 + o * 96;
    float acc = bg1[o] + bg2[o];
    for (int c = 0; c < CONV_; ++c) {
      int p = c * L + l;
      acc += (w1[c] + w2[c]) * sl[p]
           + w1[32 + c] * sl[1 * n + p] + w1[64 + c] * sl[2 * n + p]
           + w2[32 + c] * sl[3 * n + p] + w2[64 + c] * sl[4 * n + p];
    }
    x[(((size_t)b * CONV_ + o) * N_ + v) * RF_ + (RF_ - L) + l] -= 0.25f * acc;
  }
}

// ----------------------------- layernorm stats -----------------------------
__global__ void k_lnstats(const float* __restrict__ x, float* __restrict__ mu,
                          float* __restrict__ rs) {
  __shared__ float ss[256], sq[256];
  int b = blockIdx.x, tid = threadIdx.x;
  float s = 0.f, q = 0.f;
  const int TOT = CONV_ * N_ * CIN_;           // 32768
  for (int i = tid; i < TOT; i += 256) {
    int t = i & 1, v = (i >> 1) & (N_ - 1), c = i >> 10;
    float val = x[(((size_t)b * CONV_ + c) * N_ + v) * RF_ + (RF_ - CIN_) + t];
    s += val; q += val * val;
  }
  ss[tid] = s; sq[tid] = q;
  __syncthreads();
  for (int off = 128; off > 0; off >>= 1) {
    if (tid < off) { ss[tid] += ss[tid + off]; sq[tid] += sq[tid + off]; }
    __syncthreads();
  }
  if (tid == 0) {
    float m = ss[0] / (float)TOT;
    float var = sq[0] / (float)TOT - m * m;
    mu[b] = m;
    rs[b] = rsqrtf(var + 1e-5f);
  }
}

// --------------------------- heads: bc / fc MLPs ---------------------------
__global__ void k_heads(const float* __restrict__ x, const float* __restrict__ mu,
                        const float* __restrict__ rs,
                        const float* __restrict__ Wb1, const float* __restrict__ bb1,
                        const float* __restrict__ Wb2, const float* __restrict__ bb2,
                        const float* __restrict__ Wc1, const float* __restrict__ bc1,
                        const float* __restrict__ Wc2, const float* __restrict__ bc2,
                        float* __restrict__ bkc, float* __restrict__ out,
                        const float* __restrict__ ewp, int e) {
  __shared__ float xn[CONV_ * 2];
  __shared__ float hb[END_ * 2];
  __shared__ float hf[END_ * 2];
  int bv = blockIdx.x;
  int v = bv & (N_ - 1);
  int b = bv >> 9;
  int tid = threadIdx.x;                 // 128 threads
  if (tid < CONV_ * 2) {
    int c = tid >> 1, t = tid & 1;
    float val = x[(((size_t)b * CONV_ + c) * N_ + v) * RF_ + (RF_ - CIN_) + t];
    xn[c * 2 + t] = (val - mu[b]) * rs[b];
  }
  __syncthreads();
  {
    int j = tid >> 1, t = tid & 1;
    float ab = bb1[j], af = bc1[j];
    for (int c = 0; c < CONV_; ++c) {
      float xv = xn[c * 2 + t];
      ab += Wb1[j * CONV_ + c] * xv;
      af += Wc1[j * CONV_ + c] * xv;
    }
    hb[tid] = ab > 0.f ? ab : 0.f;
    hf[tid] = af > 0.f ? af : 0.f;
  }
  __syncthreads();
  if (tid < SEQ_ * 2) {                  // bc head: update residual
    int s = tid >> 1, t = tid & 1;
    float a = bb2[s];
    for (int j = 0; j < END_; ++j) a += Wb2[s * END_ + j] * hb[j * 2 + t];
    bkc[(((size_t)b * SEQ_ + s) * N_ + v) * CIN_ + t] -= a;
  } else if (tid >= 64 && tid < 64 + SEQ_) {  // fc head: last time step only
    int s = tid - 64;
    float a = bc2[s];
    for (int j = 0; j < END_; ++j) a += Wc2[s * END_ + j] * hf[j * 2 + 1];
    out[((size_t)b * SEQ_ + s) * N_ + v] += ewp[e] * a;
  }
}

// ---------------------- host-side per-iteration driver ---------------------

template <int L>
static void run_ode_iter(const float* x_in, float* x, float* Hs, float* S1,
                         float* S2, float* T1, float* T2,
                         const float* An, const float* AnT,
                         const float* Wf2, const float* Wf3, const float* Wf5,
                         const float* Wf7, const float* Wg2, const float* Wg3,
                         const float* Wg5, const float* Wg7,
                         const float* bF, const float* bG,
                         const float* Wgc1, const float* Wgc2,
                         const float* bgc1, const float* bgc2,
                         hipStream_t stream) {
  k_inception<L><<<B_ * N_, 256, 0, stream>>>(x_in, Hs, Wf2, Wf3, Wf5, Wf7,
                                              Wg2, Wg3, Wg5, Wg7, bF, bG);
  dim3 gg(L, N_ / 32);
  k_gprop<L><<<gg, 256, 0, stream>>>(An,  Hs, S1);
  k_gprop<L><<<gg, 256, 0, stream>>>(An,  S1, S2);
  k_gprop<L><<<gg, 256, 0, stream>>>(AnT, Hs, T1);
  k_gprop<L><<<gg, 256, 0, stream>>>(AnT, T1, T2);
  k_mix<L><<<B_ * N_, 256, 0, stream>>>(x, Hs, S1, S2, T1, T2,
                                        Wgc1, Wgc2, bgc1, bgc2);
}

extern "C" void kernel_launch(void* const* d_in, const int* in_sizes, int n_in,
                              void* d_out, int out_size, void* d_ws, size_t ws_size,
                              hipStream_t stream) {
  (void)in_sizes; (void)n_in; (void)out_size; (void)ws_size;
  const float* X    = (const float*)d_in[0];
  const float* adj  = (const float*)d_in[1];
  const float* ew   = (const float*)d_in[2];
  const float* Wst  = (const float*)d_in[3];
  const float* bst  = (const float*)d_in[4];
  const float* Wf2  = (const float*)d_in[5];
  const float* Wf3  = (const float*)d_in[6];
  const float* Wf5  = (const float*)d_in[7];
  const float* Wf7  = (const float*)d_in[8];
  const float* bF   = (const float*)d_in[9];
  const float* Wg2  = (const float*)d_in[10];
  const float* Wg3  = (const float*)d_in[11];
  const float* Wg5  = (const float*)d_in[12];
  const float* Wg7  = (const float*)d_in[13];
  const float* bG   = (const float*)d_in[14];
  const float* Wgc1 = (const float*)d_in[15];
  const float* bgc1 = (const float*)d_in[16];
  const float* Wgc2 = (const float*)d_in[17];
  const float* bgc2 = (const float*)d_in[18];
  const float* Wbc1 = (const float*)d_in[19];
  const float* bbc1 = (const float*)d_in[20];
  const float* Wbc2 = (const float*)d_in[21];
  const float* bbc2 = (const float*)d_in[22];
  const float* Wfc1 = (const float*)d_in[23];
  const float* bfc1 = (const float*)d_in[24];
  const float* Wfc2 = (const float*)d_in[25];
  const float* bfc2 = (const float*)d_in[26];
  float* out = (float*)d_out;

  // workspace layout (floats)
  float* ws = (float*)d_ws;
  size_t off = 0;
  auto alloc = [&](size_t nf) { float* p = ws + off; off += (nf + 63) & ~(size_t)63; return p; };
  float* An  = alloc((size_t)N_ * N_);
  float* AnT = alloc((size_t)N_ * N_);
  float* di  = alloc(N_);
  float* mu  = alloc(64);
  float* rsg = alloc(64);
  float* bkc = alloc((size_t)B_ * SEQ_ * N_ * CIN_);       // 98304
  float* x   = alloc((size_t)B_ * CONV_ * N_ * RF_);       // 3.4M
  const size_t STG = (size_t)N_ * 256 * 20;                // node-major, Lmax=20
  float* Hs = alloc(STG);
  float* S1 = alloc(STG);
  float* S2 = alloc(STG);
  float* T1 = alloc(STG);
  float* T2 = alloc(STG);

  const int NOUT = B_ * SEQ_ * N_;                         // 49152
  k_zero<<<(NOUT + 255) / 256, 256, 0, stream>>>(out, NOUT);
  k_deg<<<N_, 256, 0, stream>>>(adj, di);
  k_an<<<(N_ * N_) / 256, 256, 0, stream>>>(adj, di, An, AnT);

  const int NBKC = B_ * SEQ_ * N_ * CIN_;
  const int NX   = B_ * CONV_ * N_ * RF_;
  for (int e = 0; e < E_; ++e) {
    k_copy<<<(NBKC + 255) / 256, 256, 0, stream>>>(X, bkc, NBKC);
    for (int k = 0; k < NB_; ++k) {
      int ek = e * NB_ + k;
      k_start<<<(NX + 255) / 256, 256, 0, stream>>>(bkc, x, Wst + ek * 64, bst + ek * 32);
      const float* wf2 = Wf2 + ek * 512;
      const float* wf3 = Wf3 + ek * 768;
      const float* wf5 = Wf5 + ek * 1280;
      const float* wf7 = Wf7 + ek * 1792;
      const float* wg2 = Wg2 + ek * 512;
      const float* wg3 = Wg3 + ek * 768;
      const float* wg5 = Wg5 + ek * 1280;
      const float* wg7 = Wg7 + ek * 1792;
      const float* bf  = bF + ek * 32;
      const float* bg  = bG + ek * 32;
      const float* wc1 = Wgc1 + ek * 3072;
      const float* wc2 = Wgc2 + ek * 3072;
      const float* bc1_ = bgc1 + ek * 32;
      const float* bc2_ = bgc2 + ek * 32;
      run_ode_iter<20>(x, x, Hs, S1, S2, T1, T2, An, AnT, wf2, wf3, wf5, wf7,
                       wg2, wg3, wg5, wg7, bf, bg, wc1, wc2, bc1_, bc2_, stream);
      run_ode_iter<14>(x, x, Hs, S1, S2, T1, T2, An, AnT, wf2, wf3, wf5, wf7,
                       wg2, wg3, wg5, wg7, bf, bg, wc1, wc2, bc1_, bc2_, stream);
      run_ode_iter<8>(x, x, Hs, S1, S2, T1, T2, An, AnT, wf2, wf3, wf5, wf7,
                      wg2, wg3, wg5, wg7, bf, bg, wc1, wc2, bc1_, bc2_, stream);
      run_ode_iter<2>(x, x, Hs, S1, S2, T1, T2, An, AnT, wf2, wf3, wf5, wf7,
                      wg2, wg3, wg5, wg7, bf, bg, wc1, wc2, bc1_, bc2_, stream);
      k_lnstats<<<B_, 256, 0, stream>>>(x, mu, rsg);
      k_heads<<<B_ * N_, 128, 0, stream>>>(
          x, mu, rsg,
          Wbc1 + ek * 2048, bbc1 + ek * 64, Wbc2 + ek * 768, bbc2 + ek * 12,
          Wfc1 + ek * 2048, bfc1 + ek * 64, Wfc2 + ek * 768, bfc2 + ek * 12,
          bkc, out, ew, e);
    }
  }
}